// NeuralNet_13262859010331
// MI455X (gfx1250) — compile-verified
//
#include <hip/hip_runtime.h>

typedef __attribute__((ext_vector_type(2))) float v2f;
typedef __attribute__((ext_vector_type(8))) float v8f;

#define NCANDS 8
#define WINDOW 100
#define DDIM   300
#define TOPK   25
#define NEGV   (-1e10f)

#define AS     304        // LDS row stride for A (floats, even for b64 loads)
#define TS     304        // LDS row stride for tokens
#define NTILE  7          // ceil(100/16)
#define TROWS  112        // token rows padded to 7*16

__device__ __forceinline__ float max8(v8f a) {
    float m01 = fmaxf(a[0], a[1]);
    float m23 = fmaxf(a[2], a[3]);
    float m45 = fmaxf(a[4], a[5]);
    float m67 = fmaxf(a[6], a[7]);
    return fmaxf(fmaxf(m01, m23), fmaxf(m45, m67));
}

__global__ __launch_bounds__(128)
void ctx_embed_kernel(const float* __restrict__ emb,         // (N, 8, 300)
                      const float* __restrict__ tok,         // (N, 100, 300)
                      const unsigned char* __restrict__ mask,// (N, 100) bool
                      const float* __restrict__ bdiag,       // (300)
                      float* __restrict__ out)               // (N, 300)
{
    extern __shared__ float smem[];
    float* ldsA    = smem;                      // 16*AS   = 4864 floats
    float* ldsTok  = smem + 16 * AS;            // 112*TS  = 34048 floats
    float* scoreL  = ldsTok + TROWS * TS;       // 128
    float* bestVal = scoreL + 128;              // 32
    float* probsL  = bestVal + 32;              // 32
    int*   bestIdx = (int*)(probsL + 32);       // 32
    float* redV    = (float*)(bestIdx + 32);    // 4
    int*   redI    = (int*)(redV + 4);          // 4

    const int n    = blockIdx.x;
    const int tid  = threadIdx.x;
    const int lane = tid & 31;
    const int wave = tid >> 5;

    // ---------- Stage A' = emb[n] * diag(B) into LDS (rows 8..15 zeroed) ----------
    {
        const float4* src = (const float4*)(emb + (size_t)n * (NCANDS * DDIM));
        for (int i = tid; i < NCANDS * (DDIM / 4); i += 128) {
            int c = i / (DDIM / 4), q = i % (DDIM / 4);
            float4 v = src[c * (DDIM / 4) + q];
            int d = q * 4;
            v.x *= bdiag[d + 0]; v.y *= bdiag[d + 1];
            v.z *= bdiag[d + 2]; v.w *= bdiag[d + 3];
            *(float4*)&ldsA[c * AS + d] = v;
        }
        float4 z = make_float4(0.f, 0.f, 0.f, 0.f);
        for (int i = tid; i < 8 * (AS / 4); i += 128)          // pad rows 8..15
            *(float4*)&ldsA[8 * AS + i * 4] = z;
        // ---------- Stage raw tokens (100x300) into LDS, rows 100..111 zeroed ------
        const float4* tsrc = (const float4*)(tok + (size_t)n * (WINDOW * DDIM));
        for (int i = tid; i < WINDOW * (DDIM / 4); i += 128) {
            int w = i / (DDIM / 4), q = i % (DDIM / 4);
            *(float4*)&ldsTok[w * TS + q * 4] = tsrc[w * (DDIM / 4) + q];
        }
        for (int i = tid; i < 12 * (TS / 4); i += 128)
            *(float4*)&ldsTok[WINDOW * TS + i * 4] = z;
        if (tid < 16) scoreL[112 + tid] = NEGV;                // pad score slots
    }
    __syncthreads();

    // ---------- Scores via V_WMMA_F32_16X16X4_F32: D(16x16) = A(16x4) x B(4x16) ----
    // A frag layout: lanes 0..15 -> M=lane, K = k0,k0+1 ; lanes 16..31 -> M=lane-16, K = k0+2,k0+3
    // B frag layout: lanes 0..15 -> N=lane, K = k0,k0+1 ; lanes 16..31 -> N=lane-16, K = k0+2,k0+3
    const int mrow = lane & 15;
    const int koff = (lane >> 4) << 1;          // 0 or 2
    const int t0 = wave;                        // tiles: wave, wave+4
    const int t1 = wave + 4;
    const int t1c = (t1 > NTILE - 1) ? (NTILE - 1) : t1;   // wave3 recomputes tile6, result dropped

    const float* aRow  = ldsA   + mrow * AS + koff;
    const float* bRow0 = ldsTok + (t0 * 16 + mrow) * TS + koff;
    const float* bRow1 = ldsTok + (t1c * 16 + mrow) * TS + koff;

    v8f acc0 = {};
    v8f acc1 = {};
#pragma unroll 5
    for (int k0 = 0; k0 < DDIM; k0 += 4) {
        v2f a  = *(const v2f*)(aRow  + k0);
        v2f b0 = *(const v2f*)(bRow0 + k0);
        v2f b1 = *(const v2f*)(bRow1 + k0);
        acc0 = __builtin_amdgcn_wmma_f32_16x16x4_f32(false, a, false, b0,
                                                     (short)0, acc0, false, false);
        acc1 = __builtin_amdgcn_wmma_f32_16x16x4_f32(false, a, false, b1,
                                                     (short)0, acc1, false, false);
    }

    // ---------- token_scores = max over candidates (VGPRs 0..7, lanes 0..15) --------
    if (lane < 16) {
        float m0 = max8(acc0);
        int w = t0 * 16 + lane;
        unsigned char mk = (w < WINDOW) ? mask[(size_t)n * WINDOW + w] : (unsigned char)0;
        scoreL[w] = mk ? m0 : NEGV;
        if (t1 <= NTILE - 1) {
            float m1 = max8(acc1);
            int w1 = t1 * 16 + lane;
            unsigned char mk1 = (w1 < WINDOW) ? mask[(size_t)n * WINDOW + w1] : (unsigned char)0;
            scoreL[w1] = mk1 ? m1 : NEGV;
        }
    }
    __syncthreads();

    // ---------- top-25 by iterative argmax (128 entries, lower-index tie-break) -----
    for (int it = 0; it < TOPK; ++it) {
        float v = scoreL[tid];
        int   bi = tid;
        for (int off = 16; off > 0; off >>= 1) {
            float ov = __shfl_down(v, off, 32);
            int   oi = __shfl_down(bi, off, 32);
            if (ov > v || (ov == v && oi < bi)) { v = ov; bi = oi; }
        }
        if (lane == 0) { redV[wave] = v; redI[wave] = bi; }
        __syncthreads();
        if (tid == 0) {
            float bv = redV[0]; int bb = redI[0];
            for (int q = 1; q < 4; ++q)
                if (redV[q] > bv || (redV[q] == bv && redI[q] < bb)) { bv = redV[q]; bb = redI[q]; }
            bestVal[it] = bv;
            bestIdx[it] = bb;
            scoreL[bb]  = -3.0e38f;     // remove from further consideration
        }
        __syncthreads();
    }

    // ---------- softmax over the 25 selected scores (descending -> [0] is max) ------
    if (tid == 0) {
        float mx = bestVal[0];
        float s = 0.f;
        for (int i = 0; i < TOPK; ++i) {
            float e = __expf(bestVal[i] - mx);
            probsL[i] = e;
            s += e;
        }
        float inv = 1.f / s;
        for (int i = 0; i < TOPK; ++i) probsL[i] *= inv;
    }
    __syncthreads();

    // ---------- context = sum_i probs[i] * raw_token[bestIdx[i]]  (tokens from LDS) --
    for (int d = tid; d < DDIM; d += 128) {
        float acc = 0.f;
#pragma unroll
        for (int i = 0; i < TOPK; ++i)
            acc += probsL[i] * ldsTok[bestIdx[i] * TS + d];
        out[(size_t)n * DDIM + d] = acc;
    }
}

extern "C" void kernel_launch(void* const* d_in, const int* in_sizes, int n_in,
                              void* d_out, int out_size, void* d_ws, size_t ws_size,
                              hipStream_t stream) {
    // setup_inputs order: n, embeddingss, token_embeddingss, token_maskss, B_diag2, k
    const float*         emb   = (const float*)d_in[1];
    const float*         tok   = (const float*)d_in[2];
    const unsigned char* mask  = (const unsigned char*)d_in[3];   // jnp.bool_ -> 1 byte
    const float*         bdiag = (const float*)d_in[4];
    float*               out   = (float*)d_out;

    const int n = in_sizes[1] / (NCANDS * DDIM);   // 8192

    const size_t shmem =
        (size_t)(16 * AS + TROWS * TS + 128 + 32 + 32 + 32 + 8) * sizeof(float);

    ctx_embed_kernel<<<n, 128, shmem, stream>>>(emb, tok, mask, bdiag, out);
}